// TimeMaePreProcessing_66924180407138
// MI455X (gfx1250) — compile-verified
//
#include <hip/hip_runtime.h>
#include <hip/hip_bf16.h>

typedef __attribute__((ext_vector_type(2))) float v2f;
typedef __attribute__((ext_vector_type(8))) float v8f;

#define NB 128          // batch
#define NW 512          // num windows (T/WS)
#define KD 128          // WS*F (GEMM K)
#define EE 256          // embed dim (GEMM N)
#define NUM_KEEP 205
#define NUM_MASK 307
#define VIS_GROUPS 13   // ceil(205/16)
#define TGT_GROUPS 20   // ceil(307/16)
#define GROUPS (VIS_GROUPS + TGT_GROUPS)

// flat float offsets into d_out (tuple concatenated in return order)
#define VIS_OFF  ((size_t)0)
#define TGT_OFF  ((size_t)NB * NUM_KEEP * EE)               //  6,717,440
#define MSK_OFF  (TGT_OFF + (size_t)NB * NUM_MASK * EE)     // 16,777,216
#define VIDX_OFF (MSK_OFF + (size_t)NB * NUM_MASK * EE)     // 26,836,992
#define MIDX_OFF (VIDX_OFF + (size_t)NB * NUM_KEEP)         // 26,863,232

#define APITCH 132      // 128 + 4-float pad: 64-bank-conflict-free b64 reads

__global__ __launch_bounds__(256)
void timemae_emb_gemm(const float* __restrict__ temps,
                      const float* __restrict__ W,
                      const float* __restrict__ bias,
                      const float* __restrict__ mask_token,
                      const float* __restrict__ pos_embed,
                      const int*   __restrict__ perm,
                      float*       __restrict__ out)
{
    __shared__ float sA[16 * APITCH];
    __shared__ int   s_w[16];

    const int bid = blockIdx.x;
    const int n   = bid / GROUPS;
    const int g   = bid % GROUPS;
    const bool is_target = (g >= VIS_GROUPS);
    const int slot_base  = is_target ? (g - VIS_GROUPS) * 16 : g * 16;
    const int limit      = is_target ? NUM_MASK : NUM_KEEP;
    int valid = limit - slot_base; if (valid > 16) valid = 16;

    const int tid = threadIdx.x;

    // gather the 16 window ids for this output group (clamp tail -> dup rows)
    if (tid < 16) {
        int slot = slot_base + tid;
        if (slot > limit - 1) slot = limit - 1;
        const int pcol = is_target ? (NUM_KEEP + slot) : slot;
        s_w[tid] = perm[n * NW + pcol];
    }
    __syncthreads();

    // stage A tile (16 gathered windows x 128 K) into LDS; 8 floats/thread
    {
        const int row = tid >> 4;
        const int col = (tid & 15) * 8;
        const float* src = temps + ((size_t)n * NW + (size_t)s_w[row]) * KD + col;
        float4 x = *(const float4*)(src);
        float4 y = *(const float4*)(src + 4);
        *(float4*)(&sA[row * APITCH + col])     = x;
        *(float4*)(&sA[row * APITCH + col + 4]) = y;
    }
    __syncthreads();

    const int wv   = tid >> 5;      // wave id: 8 waves
    const int lane = tid & 31;
    const int hf   = lane >> 4;     // lane half
    const int lm   = lane & 15;

    const int n0a = wv * 32;        // this wave's two 16-wide N tiles
    const int n0b = n0a + 16;

    // B: col N = n0+lm, K pair (kk, kk+1) == contiguous float2 in row-major W
    const float* Wa   = W + (size_t)(n0a + lm) * KD;
    const float* Wb   = W + (size_t)(n0b + lm) * KD;
    const float* Arow = &sA[lm * APITCH];

    v8f acc0 = {};
    v8f acc1 = {};
    #pragma unroll
    for (int k0 = 0; k0 < KD; k0 += 4) {
        const int kk = k0 + 2 * hf;
        v2f a  = *(const v2f*)(Arow + kk);
        v2f b0 = *(const v2f*)(Wa + kk);
        v2f b1 = *(const v2f*)(Wb + kk);
        acc0 = __builtin_amdgcn_wmma_f32_16x16x4_f32(false, a, false, b0,
                                                     (short)0, acc0, false, false);
        acc1 = __builtin_amdgcn_wmma_f32_16x16x4_f32(false, a, false, b1,
                                                     (short)0, acc1, false, false);
    }

    // epilogue: + bias + pos_embed[w]; target path also emits masked_input
    const float bias0 = bias[n0a + lm];
    const float bias1 = bias[n0b + lm];
    float mt0 = 0.f, mt1 = 0.f;
    if (is_target) { mt0 = mask_token[n0a + lm]; mt1 = mask_token[n0b + lm]; }

    float* embBase = out + (is_target ? TGT_OFF : VIS_OFF);
    float* mskBase = out + MSK_OFF;
    const int rows_total = is_target ? NUM_MASK : NUM_KEEP;

    #pragma unroll
    for (int r = 0; r < 8; ++r) {
        const int M = r + 8 * hf;   // C/D layout: VGPR r -> row r (+8 for hi half)
        if (M < valid) {
            const int w = s_w[M];
            const float pe0 = pos_embed[(size_t)w * EE + n0a + lm];
            const float pe1 = pos_embed[(size_t)w * EE + n0b + lm];
            const size_t o = ((size_t)n * rows_total + (size_t)(slot_base + M)) * EE;
            embBase[o + n0a + lm] = acc0[r] + bias0 + pe0;
            embBase[o + n0b + lm] = acc1[r] + bias1 + pe1;
            if (is_target) {
                mskBase[o + n0a + lm] = mt0 + pe0;
                mskBase[o + n0b + lm] = mt1 + pe1;
            }
        }
    }
}

__global__ __launch_bounds__(256)
void timemae_indices(const int* __restrict__ perm, float* __restrict__ out)
{
    const int i = blockIdx.x * blockDim.x + threadIdx.x;
    if (i >= NB * NW) return;
    const int n = i / NW;
    const int j = i % NW;
    const float v = (float)perm[i];   // window ids < 512: exact in fp32
    if (j < NUM_KEEP) out[VIDX_OFF + (size_t)n * NUM_KEEP + j] = v;
    else              out[MIDX_OFF + (size_t)n * NUM_MASK + (j - NUM_KEEP)] = v;
}

extern "C" void kernel_launch(void* const* d_in, const int* in_sizes, int n_in,
                              void* d_out, int out_size, void* d_ws, size_t ws_size,
                              hipStream_t stream)
{
    const float* temps      = (const float*)d_in[0];
    const float* W          = (const float*)d_in[1];
    const float* b          = (const float*)d_in[2];
    const float* mask_token = (const float*)d_in[3];
    const float* pos_embed  = (const float*)d_in[4];
    const int*   perm       = (const int*)  d_in[5];
    float* out = (float*)d_out;

    timemae_emb_gemm<<<dim3(NB * GROUPS), dim3(256), 0, stream>>>(
        temps, W, b, mask_token, pos_embed, perm, out);
    timemae_indices<<<dim3((NB * NW + 255) / 256), dim3(256), 0, stream>>>(perm, out);
}